// Fasttext_25512105738892
// MI455X (gfx1250) — compile-verified
//
#include <hip/hip_runtime.h>
#include <hip/hip_bf16.h>

// FastText embed-gather + projection as a skinny fp32 GEMM on CDNA5 WMMA.
// out[tok, p] = sum_e embed[ids[tok]][e] * W[p][e] + b[p], masked by seq_len.
//
// Per wave (wave32): one 16-token x 128-proj tile.
//   A = gathered embed rows, 16x4 fp32 per WMMA step (v2f per lane)
//   B = W^T chunk,            4x16 fp32 per WMMA step (v2f per lane)
//   C = 8 x (16x16 fp32)  accumulated over 75 K-steps of 4 (K=300)
// using v_wmma_f32_16x16x4_f32.

typedef __attribute__((ext_vector_type(2))) float v2f;
typedef __attribute__((ext_vector_type(8))) float v8f;

#define EMBED 300
#define PROJ  128
#define SEQ   2048
#define KSTEPS (EMBED / 4)       // 75
#define NTILES (PROJ / 16)       // 8
#define WAVES_PER_BLOCK 8

__global__ __launch_bounds__(256) void fasttext_wmma_kernel(
    const int*   __restrict__ ids,      // [B*S] flattened token ids
    const int*   __restrict__ lens,     // [B] sequence lengths
    const float* __restrict__ embed,    // [VOCAB, 300]
    const float* __restrict__ W,        // [128, 300] row-major
    const float* __restrict__ bias,     // [128]
    float*       __restrict__ out)      // [B*S, 128]
{
    const int lane = threadIdx.x & 31;
    const int wave = threadIdx.x >> 5;
    const int tile = blockIdx.x * WAVES_PER_BLOCK + wave; // 2048 tiles total
    const int tok_base = tile * 16;

    const int half = lane >> 4;   // 0: K offsets {0,1}, 1: K offsets {2,3}
    const int l16  = lane & 15;   // M (for A) / N (for B) index

    // ---- A operand: gathered embedding row for token m = l16 -------------
    // A[m][k] layout: lane = m + 16*(k>=2), vgpr v -> k = 2*half + v.
    const int tokA = tok_base + l16;
    const long idA = (long)ids[tokA];
    const float* arow = embed + idA * EMBED + 2 * half; // float2 per K-step

    // ---- B operand: W^T chunks. B[k][n] = W[n][k]. ------------------------
    // B layout mirrors A: lane -> n = l16, k = 2*half + v  => contiguous
    // float2 from W[(ntile*16 + l16)*300 + kbase + 2*half].
    const float* wrow = W + (long)l16 * EMBED + 2 * half;

    // ---- accumulate ------------------------------------------------------
    v8f c[NTILES];
#pragma unroll
    for (int j = 0; j < NTILES; ++j) c[j] = (v8f)0.0f;

    for (int k = 0; k < KSTEPS; ++k) {
        const v2f a = *(const v2f*)(arow + k * 4);
#pragma unroll
        for (int j = 0; j < NTILES; ++j) {
            const v2f bm = *(const v2f*)(wrow + (long)j * (16 * EMBED) + k * 4);
            // 8-arg fp32 WMMA: (neg_a, A, neg_b, B, c_mod, C, reuse_a, reuse_b)
            c[j] = __builtin_amdgcn_wmma_f32_16x16x4_f32(
                false, a, false, bm, (short)0, c[j], false, false);
        }
    }

    // ---- bias (per lane: one value per N-tile) ---------------------------
    float bv[NTILES];
#pragma unroll
    for (int j = 0; j < NTILES; ++j) bv[j] = bias[j * 16 + l16];

    // ---- mask: one sentence per tile (2048 % 16 == 0) --------------------
    const int blen = lens[tok_base >> 11];  // tok / 2048
    const int s0   = tok_base & (SEQ - 1);

    // ---- store: C/D layout m = r + 8*half, n = l16 -----------------------
#pragma unroll
    for (int r = 0; r < 8; ++r) {
        const int m     = r + 8 * half;
        const int tok   = tok_base + m;
        const bool keep = (s0 + m) < blen;
        float* orow = out + (long)tok * PROJ + l16;
#pragma unroll
        for (int j = 0; j < NTILES; ++j) {
            const float v = c[j][r] + bv[j];
            orow[j * 16] = keep ? v : 0.0f;  // poisoned d_out: write zeros
        }
    }
}

extern "C" void kernel_launch(void* const* d_in, const int* in_sizes, int n_in,
                              void* d_out, int out_size, void* d_ws, size_t ws_size,
                              hipStream_t stream) {
    (void)in_sizes; (void)n_in; (void)d_ws; (void)ws_size; (void)out_size;
    const int*   ids   = (const int*)d_in[0];   // ext_word_ids [16,2048]
    const int*   lens  = (const int*)d_in[1];   // seq_lengths  [16]
    const float* embed = (const float*)d_in[2]; // [200000,300]
    const float* W     = (const float*)d_in[3]; // [128,300]
    const float* bias  = (const float*)d_in[4]; // [128]
    float*       out   = (float*)d_out;         // [16,2048,128]

    // 32768 tokens / 16 per tile = 2048 tiles; 8 waves (256 thr) per block.
    const int blocks = 2048 / WAVES_PER_BLOCK;  // 256
    fasttext_wmma_kernel<<<blocks, 256, 0, stream>>>(ids, lens, embed, W, bias, out);
}